// SuperLoRAGroup_85847806312581
// MI455X (gfx1250) — compile-verified
//
#include <hip/hip_runtime.h>
#include <hip/hip_bf16.h>

// ---------------------------------------------------------------------------
// SuperLoRAGroup fastfood pipeline for gfx1250 (MI455X)
//   delta = (Bw @ A) * (alpha/rank)           -- fp32 WMMA 16x16x4
//   x     = pad(delta.flat) * sign
//   x     = FWHT_2^24(x)                      -- H4096 (x) H4096 split passes
//   y[i]  = x[perm[i]] * G[i]
//   out   = FWHT_2^24(y)
// ---------------------------------------------------------------------------

#define GROUP_SIZE  16777216   // 2^24
#define INNER       4194304    // 2^22
#define PADDED      16777216   // 2^24
#define D_DIM       2048
#define RANK        16
#define SCALING_F   (1.0f / 16.0f)          // alpha / rank
#define FWHT_NORM   (1.0f / 4096.0f)        // 1/sqrt(2^24) per full transform

typedef __attribute__((ext_vector_type(2))) float v2f;
typedef __attribute__((ext_vector_type(8))) float v8f;

// ---------------------------------------------------------------------------
// Kernel 1a: zero the pad tail X[INNER .. PADDED)
// ---------------------------------------------------------------------------
__global__ void sl_zero_tail(float* __restrict__ X) {
    unsigned t = blockIdx.x * blockDim.x + threadIdx.x;   // 3,145,728 threads
    float4* p = (float4*)(X + INNER);
    float4 z = {0.f, 0.f, 0.f, 0.f};
    p[t] = z;
}

// ---------------------------------------------------------------------------
// Kernel 1b: WMMA GEMM  delta = Bw(2048x16) @ A(16x2048), fused * scale * sign
// One wave per 16x16 output tile; K=16 as 4 chained V_WMMA_F32_16X16X4_F32.
//   A-operand (16x4 f32, 2 VGPR): lanes 0-15 M=lane K={kb,kb+1};
//                                 lanes 16-31 M=lane-16 K={kb+2,kb+3}
//   B-operand (4x16 f32, 2 VGPR): symmetric, N striped across lanes
//   C/D (16x16 f32, 8 VGPR): vj -> M = j + 8*(lane>>4), N = lane&15
// ---------------------------------------------------------------------------
__global__ void sl_gemm_wmma(const float* __restrict__ Amat,   // (RANK, D) row-major
                             const float* __restrict__ Bw,     // (D, RANK) row-major
                             const float* __restrict__ sign,
                             float* __restrict__ X) {
    const unsigned gtid = blockIdx.x * blockDim.x + threadIdx.x;
    const unsigned wave = gtid >> 5;           // 0 .. 16383
    const unsigned lane = threadIdx.x & 31;
    const unsigned tm = wave >> 7;             // tile row 0..127
    const unsigned tn = wave & 127;            // tile col 0..127
    const unsigned m0 = tm * 16, n0 = tn * 16;
    const unsigned half = lane >> 4;           // 0: K pair {0,1}, 1: K pair {2,3}
    const unsigned ln = lane & 15;

    v8f c = {};
    #pragma unroll
    for (int kk = 0; kk < 4; ++kk) {
        const unsigned kb = kk * 4 + half * 2;
        v2f a, b;
        a.x = Bw[(m0 + ln) * RANK + kb];
        a.y = Bw[(m0 + ln) * RANK + kb + 1];
        b.x = Amat[kb * D_DIM + n0 + ln];
        b.y = Amat[(kb + 1) * D_DIM + n0 + ln];
        c = __builtin_amdgcn_wmma_f32_16x16x4_f32(
                /*neg_a=*/false, a, /*neg_b=*/false, b,
                /*c_mod=*/(short)0, c, /*reuse_a=*/false, /*reuse_b=*/false);
    }

    #pragma unroll
    for (int j = 0; j < 8; ++j) {
        const unsigned row = m0 + j + half * 8;
        const unsigned col = n0 + ln;
        const unsigned i = row * (unsigned)D_DIM + col;   // flat index < INNER
        X[i] = c[j] * SCALING_F * sign[i];
    }
}

// ---------------------------------------------------------------------------
// Kernel 2: FWHT over low 12 bits — contiguous 4096-float block per workgroup,
// 12 butterfly stages in 16 KB static LDS. In place.
// ---------------------------------------------------------------------------
__global__ void sl_fwht_row(float* __restrict__ X) {
    __shared__ float s[4096];
    const unsigned tid = threadIdx.x;                 // 256 threads
    float4* gv = (float4*)(X + (size_t)blockIdx.x * 4096);
    float4* sv = (float4*)s;

    #pragma unroll
    for (int q = 0; q < 4; ++q)
        sv[tid + 256 * q] = gv[tid + 256 * q];
    __syncthreads();

    #pragma unroll
    for (int st = 0; st < 12; ++st) {
        const unsigned step = 1u << st;
        #pragma unroll
        for (int q = 0; q < 8; ++q) {                 // 2048 pairs / 256 threads
            const unsigned p  = tid + 256 * q;
            const unsigned i0 = ((p >> st) << (st + 1)) | (p & (step - 1));
            const unsigned i1 = i0 + step;
            const float a = s[i0], b = s[i1];
            s[i0] = a + b;
            s[i1] = a - b;
        }
        __syncthreads();
    }

    #pragma unroll
    for (int q = 0; q < 4; ++q)
        gv[tid + 256 * q] = sv[tid + 256 * q];
}

// ---------------------------------------------------------------------------
// Kernel 3: FWHT over high 12 bits — view X as 4096x4096 row-major; each
// workgroup owns 16 columns (4096 x 16 slab), held in 272 KB dynamic LDS
// (stride 17 padding kills bank conflicts).  CDNA5's 320KB/WGP LDS makes this
// single-pass slab possible.  Applies the 1/sqrt(2^24) normalization on store.
// ---------------------------------------------------------------------------
__global__ void sl_fwht_col(float* __restrict__ X, float norm) {
    extern __shared__ float s[];                      // 4096 * 17 floats
    const unsigned tid = threadIdx.x;                 // 1024 threads
    const unsigned cbase = blockIdx.x * 16;

    #pragma unroll
    for (int q = 0; q < 64; ++q) {                    // 65536 elems / 1024 thr
        const unsigned idx = tid + 1024 * q;
        const unsigned r = idx >> 4, c = idx & 15;
        s[r * 17 + c] = X[(r << 12) + cbase + c];
    }
    __syncthreads();

    #pragma unroll
    for (int st = 0; st < 12; ++st) {
        const unsigned step = 1u << st;
        #pragma unroll
        for (int q = 0; q < 32; ++q) {                // 32768 pairs / 1024 thr
            const unsigned pt  = tid + 1024 * q;
            const unsigned col = pt & 15;
            const unsigned p   = pt >> 4;             // pair within column
            const unsigned i0  = ((p >> st) << (st + 1)) | (p & (step - 1));
            const unsigned a0  = i0 * 17 + col;
            const unsigned a1  = (i0 + step) * 17 + col;
            const float a = s[a0], b = s[a1];
            s[a0] = a + b;
            s[a1] = a - b;
        }
        __syncthreads();
    }

    #pragma unroll
    for (int q = 0; q < 64; ++q) {
        const unsigned idx = tid + 1024 * q;
        const unsigned r = idx >> 4, c = idx & 15;
        X[(r << 12) + cbase + c] = s[r * 17 + c] * norm;
    }
}

// ---------------------------------------------------------------------------
// Kernel 4: permutation gather  out[i] = X[perm[i]] * G[i]
// ---------------------------------------------------------------------------
__global__ void sl_perm_gather(const float* __restrict__ X,
                               const int* __restrict__ perm,
                               const float* __restrict__ G,
                               float* __restrict__ out) {
    const unsigned base = blockIdx.x * 1024 + threadIdx.x;
    #pragma unroll
    for (int q = 0; q < 4; ++q) {
        const unsigned i = base + 256 * q;
        out[i] = X[perm[i]] * G[i];
    }
}

// ---------------------------------------------------------------------------
extern "C" void kernel_launch(void* const* d_in, const int* in_sizes, int n_in,
                              void* d_out, int out_size, void* d_ws, size_t ws_size,
                              hipStream_t stream) {
    const float* Amat = (const float*)d_in[0];   // (16, 2048)
    const float* Bw   = (const float*)d_in[1];   // (2048, 16)
    const float* G    = (const float*)d_in[2];   // (2^24,)
    const float* sign = (const float*)d_in[3];   // (2^24,)
    const int*   perm = (const int*)d_in[4];     // (2^24,)
    float* out = (float*)d_out;
    float* X   = (float*)d_ws;                   // 64 MB scratch

    const size_t colLds = (size_t)4096 * 17 * sizeof(float);   // 278,528 B

    // 1) x = pad(delta) * sign
    sl_zero_tail <<< 12288, 256, 0, stream >>>(X);             // (PADDED-INNER)/4 elems
    sl_gemm_wmma <<< 2048, 256, 0, stream >>>(Amat, Bw, sign, X); // 16384 waves

    // 2) x = FWHT(x)
    sl_fwht_row  <<< 4096, 256, 0, stream >>>(X);
    sl_fwht_col  <<< 256, 1024, colLds, stream >>>(X, FWHT_NORM);

    // 3) y = x[perm] * G  -> out
    sl_perm_gather <<< 16384, 256, 0, stream >>>(X, perm, G, out);

    // 4) out = FWHT(out)
    sl_fwht_row  <<< 4096, 256, 0, stream >>>(out);
    sl_fwht_col  <<< 256, 1024, colLds, stream >>>(out, FWHT_NORM);
}